// EigenModule_38482906972748
// MI455X (gfx1250) — compile-verified
//
#include <hip/hip_runtime.h>
#include <math.h>

#define NB    16
#define N     1024
#define KSUB  32
#define NLOW  16
#define NITER 300
#define VPAD  1028   // LDS column stride (floats): (4*col + k) % 64 tiles all banks

typedef __attribute__((ext_vector_type(2))) float v2f;
typedef __attribute__((ext_vector_type(8))) float v8f;

// ---------------------------------------------------------------------------
// Init: zero per-batch bound accumulators, fill V with deterministic pseudo-
// random values (ws is poisoned 0xAA, so everything must be initialized).
// ---------------------------------------------------------------------------
__global__ void init_kernel(float* __restrict__ V, float* __restrict__ gersh,
                            float* __restrict__ fro2) {
  int e = blockIdx.x * blockDim.x + threadIdx.x;
  if (e < NB) { gersh[e] = 0.0f; fro2[e] = 0.0f; }
  const int total = NB * N * KSUB;
  for (int i = e; i < total; i += gridDim.x * blockDim.x) {
    unsigned h = (unsigned)i * 2654435761u;
    h ^= h >> 16; h *= 2246822519u; h ^= h >> 13;
    V[i] = (float)(h & 0xFFFFu) * (1.0f / 65536.0f) - 0.5f;
  }
}

// ---------------------------------------------------------------------------
// Spectral upper bound accumulators: Gershgorin row-sum (positive-float uint
// atomicMax) and Frobenius norm^2 (atomicAdd), per batch.
// ---------------------------------------------------------------------------
__global__ void bound_kernel(const float* __restrict__ A, float* __restrict__ gersh,
                             float* __restrict__ fro2) {
  int t = blockIdx.x * blockDim.x + threadIdx.x;   // NB*N threads
  int b = t >> 10, i = t & (N - 1);
  const float* Ab = A + (size_t)b * N * N;
  float g = 0.0f, f = 0.0f;
  for (int j = 0; j < N; ++j) {
    float m = 0.5f * (Ab[(size_t)i * N + j] + Ab[(size_t)j * N + i]);
    g += fabsf(m);
    f += m * m;
  }
  atomicMax((unsigned int*)(gersh + b), __float_as_uint(g));
  atomicAdd(fro2 + b, f);
}

// c_b = min(Gershgorin, 2.5*||M||_F/sqrt(N))  (upper bound on lambda_max)
__global__ void cfin_kernel(const float* __restrict__ gersh,
                            const float* __restrict__ fro2,
                            float* __restrict__ cvals) {
  int b = threadIdx.x;
  if (b < NB) cvals[b] = fminf(gersh[b], 2.5f * sqrtf(fro2[b]) * (1.0f / 32.0f));
}

// ---------------------------------------------------------------------------
// One-time build of the shifted operator: Ms = c*I - 0.5*(A + A^T).
// Lowest eigenpairs of M are the dominant eigenpairs of Ms.  64 MB, L2-fits.
// ---------------------------------------------------------------------------
__global__ void symm_kernel(const float* __restrict__ A,
                            const float* __restrict__ cvals,
                            float* __restrict__ Ms) {
  size_t t = (size_t)blockIdx.x * blockDim.x + threadIdx.x;
  const size_t total = (size_t)NB * N * N;
  for (size_t e = t; e < total; e += (size_t)gridDim.x * blockDim.x) {
    size_t b = e >> 20;
    int    i = (int)((e >> 10) & (N - 1));
    int    j = (int)(e & (N - 1));
    const float* Ab = A + b * N * N;
    float v = -0.5f * (Ab[(size_t)i * N + j] + Ab[(size_t)j * N + i]);
    if (i == j) v += cvals[b];
    Ms[e] = v;
  }
}

// ---------------------------------------------------------------------------
// WMMA fp32 GEMM (pre-built operator path):  W = Ms * V.
// One wave -> 16x32 output strip.  V panel staged transposed+padded in LDS
// (bank-conflict-free ds_load_b64 B operands); A-operand loads coalesced.
// ---------------------------------------------------------------------------
__global__ __launch_bounds__(128) void gemm_pre_kernel(
    const float* __restrict__ Ms, const float* __restrict__ Vin,
    float* __restrict__ Wout) {
  extern __shared__ float Vs[];                     // KSUB * VPAD floats
  int b  = blockIdx.x >> 4;                         // 16 blocks per batch
  int mt = (blockIdx.x & 15) * 4 + (threadIdx.x >> 5);
  int lane = threadIdx.x & 31;
  int half = lane >> 4;
  int l    = lane & 15;

  // Stage V transposed: Vs[c*VPAD + k] = V[k*KSUB + c]  (coalesced global)
  {
    const float* Vp = Vin + (size_t)b * N * KSUB;
    for (int e = threadIdx.x; e < N * KSUB; e += 128) {
      int k = e >> 5, c = e & 31;
      Vs[c * VPAD + k] = Vp[e];
    }
  }
  __syncthreads();

  const float* Mb = Ms + (size_t)b * N * N;
  int m = mt * 16 + l;                              // output row (A-operand M)
  v8f acc0 = {};                                    // subspace cols 0..15
  v8f acc1 = {};                                    // subspace cols 16..31

  for (int k0 = 0; k0 < N; k0 += 4) {
    int ks = k0 + 2 * half;   // lanes 0-15: K=k0,k0+1 ; lanes 16-31: K=k0+2,k0+3
    __builtin_prefetch(Mb + (size_t)(ks + 32) * N + m, 0, 3);
    v2f a;
    a.x = Mb[(size_t)ks * N + m];                   // row-coalesced b32
    a.y = Mb[(size_t)(ks + 1) * N + m];
    v2f b0 = *(const v2f*)(Vs + l * VPAD + ks);          // ds_load_b64
    v2f b1 = *(const v2f*)(Vs + (16 + l) * VPAD + ks);   // ds_load_b64

    acc0 = __builtin_amdgcn_wmma_f32_16x16x4_f32(false, a, false, b0,
                                                 (short)0, acc0, false, false);
    acc1 = __builtin_amdgcn_wmma_f32_16x16x4_f32(false, a, false, b1,
                                                 (short)0, acc1, false, false);
  }

  float* Wb = Wout + (size_t)b * N * KSUB;
#pragma unroll
  for (int r = 0; r < 8; ++r) {
    int row = mt * 16 + r + 8 * half;
    Wb[row * KSUB + l]      = acc0[r];
    Wb[row * KSUB + 16 + l] = acc1[r];
  }
}

// ---------------------------------------------------------------------------
// WMMA fp32 GEMM (fallback, small ws): shifted operator built on the fly.
//   W = (c*I - Msym) * V
// ---------------------------------------------------------------------------
__global__ __launch_bounds__(128) void gemm_fly_kernel(
    const float* __restrict__ A, const float* __restrict__ Vin,
    float* __restrict__ Wout, const float* __restrict__ cvals) {
  int wid  = blockIdx.x * 4 + (threadIdx.x >> 5);
  int lane = threadIdx.x & 31;
  int half = lane >> 4;
  int l    = lane & 15;
  int b    = wid >> 6;
  int mt   = wid & 63;

  const float* Ab = A   + (size_t)b * N * N;
  const float* Vb = Vin + (size_t)b * N * KSUB;
  float cval = cvals[b];
  int m = mt * 16 + l;
  v8f acc0 = {};
  v8f acc1 = {};

  for (int k0 = 0; k0 < N; k0 += 4) {
    int ks = k0 + 2 * half;
    float2 tp = *(const float2*)(Ab + (size_t)m * N + ks);
    float  r0 = Ab[(size_t)ks * N + m];
    float  r1 = Ab[(size_t)(ks + 1) * N + m];
    v2f a;
    a.x = -0.5f * (r0 + tp.x) + ((m == ks    ) ? cval : 0.0f);
    a.y = -0.5f * (r1 + tp.y) + ((m == ks + 1) ? cval : 0.0f);
    v2f b0, b1;
    b0.x = Vb[ks * KSUB + l];       b0.y = Vb[(ks + 1) * KSUB + l];
    b1.x = Vb[ks * KSUB + 16 + l];  b1.y = Vb[(ks + 1) * KSUB + 16 + l];

    acc0 = __builtin_amdgcn_wmma_f32_16x16x4_f32(false, a, false, b0,
                                                 (short)0, acc0, false, false);
    acc1 = __builtin_amdgcn_wmma_f32_16x16x4_f32(false, a, false, b1,
                                                 (short)0, acc1, false, false);
  }

  float* Wb = Wout + (size_t)b * N * KSUB;
#pragma unroll
  for (int r = 0; r < 8; ++r) {
    int row = mt * 16 + r + 8 * half;
    Wb[row * KSUB + l]      = acc0[r];
    Wb[row * KSUB + 16 + l] = acc1[r];
  }
}

// ---------------------------------------------------------------------------
// CholQR orthonormalization: G = W^T W (LDS), Cholesky G = R^T R, V = W R^-1.
// One workgroup per batch.
// ---------------------------------------------------------------------------
__global__ __launch_bounds__(256) void ortho_kernel(const float* __restrict__ W,
                                                    float* __restrict__ V) {
  __shared__ float G[KSUB * KSUB];
  int b = blockIdx.x, tid = threadIdx.x;
  const float* Wb = W + (size_t)b * N * KSUB;
  float*       Vb = V + (size_t)b * N * KSUB;

  for (int e = tid; e < KSUB * KSUB; e += 256) {
    int i = e >> 5, j = e & 31;
    float s = 0.0f;
    for (int r = 0; r < N; ++r) s += Wb[r * KSUB + i] * Wb[r * KSUB + j];
    G[e] = s;
  }
  __syncthreads();

  if (tid == 0) {  // upper-triangular Cholesky in LDS
    for (int k = 0; k < KSUB; ++k) {
      float d = sqrtf(fmaxf(G[k * KSUB + k], 1e-30f));
      G[k * KSUB + k] = d;
      float inv = 1.0f / d;
      for (int j = k + 1; j < KSUB; ++j) G[k * KSUB + j] *= inv;
      for (int i = k + 1; i < KSUB; ++i)
        for (int j = i; j < KSUB; ++j)
          G[i * KSUB + j] -= G[k * KSUB + i] * G[k * KSUB + j];
    }
  }
  __syncthreads();

  for (int row = tid; row < N; row += 256) {  // solve x * R = w per row
    float x[KSUB];
    for (int j = 0; j < KSUB; ++j) {
      float xj = Wb[row * KSUB + j];
      for (int t2 = 0; t2 < j; ++t2) xj -= x[t2] * G[t2 * KSUB + j];
      x[j] = xj / G[j * KSUB + j];
    }
    for (int j = 0; j < KSUB; ++j) Vb[row * KSUB + j] = x[j];
  }
}

// ---------------------------------------------------------------------------
// Rayleigh-Ritz on the shifted operator: W = (cI - M)V, so S = V^T W =
// cI - V^T M V  (V^T V = I).  Ritz pairs of M: lambda = c - theta, same
// eigenvectors.  Wave-parallel 32x32 cyclic Jacobi (wave32 lockstep, lane r
// owns row/col r); sort lambda ascending; emit 16 lowest pairs.
// ---------------------------------------------------------------------------
__global__ __launch_bounds__(256) void rr_kernel(const float* __restrict__ V,
                                                 const float* __restrict__ W,
                                                 const float* __restrict__ cvals,
                                                 float* __restrict__ out) {
  __shared__ float S[KSUB * KSUB];
  __shared__ float U[KSUB * KSUB];
  __shared__ float lam[KSUB];
  __shared__ int   perm[KSUB];
  int b = blockIdx.x, tid = threadIdx.x;
  const float* Vb = V + (size_t)b * N * KSUB;
  const float* Wb = W + (size_t)b * N * KSUB;
  float cval = cvals[b];

  for (int e = tid; e < KSUB * KSUB; e += 256) {
    int i = e >> 5, j = e & 31;
    float s = 0.0f;
    for (int r = 0; r < N; ++r) s += Vb[r * KSUB + i] * Wb[r * KSUB + j];
    S[e] = s;
    U[e] = (i == j) ? 1.0f : 0.0f;
  }
  __syncthreads();

  if (tid < KSUB) {        // one wave32, lane r updates row/col r
    int r = tid;
    for (int sweep = 0; sweep < 24; ++sweep) {
      for (int p = 0; p < KSUB - 1; ++p) {
        for (int q = p + 1; q < KSUB; ++q) {
          float apq = S[p * KSUB + q];               // uniform across lanes
          if (fabsf(apq) > 1e-9f) {
            float app = S[p * KSUB + p], aqq = S[q * KSUB + q];
            float tau = (aqq - app) / (2.0f * apq);
            float t   = copysignf(1.0f, tau) / (fabsf(tau) + sqrtf(1.0f + tau * tau));
            float c   = 1.0f / sqrtf(1.0f + t * t);
            float sn  = t * c;
            float sp = S[p * KSUB + r], sq = S[q * KSUB + r];
            S[p * KSUB + r] = c * sp - sn * sq;      // rows p,q  (J^T S)
            S[q * KSUB + r] = sn * sp + c * sq;
            sp = S[r * KSUB + p]; sq = S[r * KSUB + q];
            S[r * KSUB + p] = c * sp - sn * sq;      // cols p,q  (S J)
            S[r * KSUB + q] = sn * sp + c * sq;
            float up = U[r * KSUB + p], uq = U[r * KSUB + q];
            U[r * KSUB + p] = c * up - sn * uq;      // U <- U J
            U[r * KSUB + q] = sn * up + c * uq;
          }
        }
      }
    }
  }
  __syncthreads();

  if (tid == 0) {          // lambda = c - theta; sort ascending
    for (int i = 0; i < KSUB; ++i) perm[i] = i;
    for (int i = 0; i < KSUB; ++i) lam[i] = cval - S[i * KSUB + i];
    for (int i = 0; i < KSUB - 1; ++i) {
      int best = i;
      for (int j = i + 1; j < KSUB; ++j)
        if (lam[perm[j]] < lam[perm[best]]) best = j;
      int tmp = perm[i]; perm[i] = perm[best]; perm[best] = tmp;
    }
  }
  __syncthreads();

  if (tid < NLOW) out[b * NLOW + tid] = lam[perm[tid]];
  float* ovec = out + NB * NLOW;                     // (B, N, NLOW)
  for (int e = tid; e < N * NLOW; e += 256) {
    int i = e >> 4, j = e & 15;
    int k = perm[j];
    float s = 0.0f;
    for (int t2 = 0; t2 < KSUB; ++t2) s += Vb[i * KSUB + t2] * U[t2 * KSUB + k];
    ovec[((size_t)b * N + i) * NLOW + j] = s;
  }
}

// ---------------------------------------------------------------------------
extern "C" void kernel_launch(void* const* d_in, const int* in_sizes, int n_in,
                              void* d_out, int out_size, void* d_ws, size_t ws_size,
                              hipStream_t stream) {
  (void)in_sizes; (void)n_in; (void)out_size;
  const float* A = (const float*)d_in[0];   // (16,1024,1024) fp32; d_in[1]=nlowest(=16) fixed
  float* base = (float*)d_ws;

  const size_t elemsM  = (size_t)NB * N * N;          // 16.8M floats (64 MB)
  const size_t elemsVW = (size_t)NB * N * KSUB;       // 512K floats each
  const size_t smallEl = (size_t)NB * 3 + 64;
  const bool pre = ws_size >= (elemsM + 2 * elemsVW + smallEl) * sizeof(float);

  float* Ms    = pre ? base : nullptr;
  float* V     = pre ? (base + elemsM) : base;
  float* W     = V + elemsVW;
  float* gersh = W + elemsVW;
  float* fro2  = gersh + NB;
  float* cvals = fro2 + NB;
  float* out   = (float*)d_out;

  init_kernel<<<256, 256, 0, stream>>>(V, gersh, fro2);
  bound_kernel<<<(NB * N) / 256, 256, 0, stream>>>(A, gersh, fro2);
  cfin_kernel<<<1, NB, 0, stream>>>(gersh, fro2, cvals);
  if (pre) symm_kernel<<<1024, 256, 0, stream>>>(A, cvals, Ms);

  const unsigned smem = KSUB * VPAD * sizeof(float);  // ~128.5 KB V panel
  for (int it = 0; it < NITER; ++it) {
    if (pre) gemm_pre_kernel<<<256, 128, smem, stream>>>(Ms, V, W);
    else     gemm_fly_kernel<<<256, 128, 0,    stream>>>(A,  V, W, cvals);
    ortho_kernel<<<NB, 256, 0, stream>>>(W, V);
  }
  // Final application (same shifted operator), then Rayleigh-Ritz extraction
  if (pre) gemm_pre_kernel<<<256, 128, smem, stream>>>(Ms, V, W);
  else     gemm_fly_kernel<<<256, 128, 0,    stream>>>(A,  V, W, cvals);
  rr_kernel<<<NB, 256, 0, stream>>>(V, W, cvals, out);
}